// linear_sub_86784109183252
// MI455X (gfx1250) — compile-verified
//
#include <hip/hip_runtime.h>

#define DIM 512
#define BATCH 65536

typedef __attribute__((ext_vector_type(2))) float v2f;
typedef __attribute__((ext_vector_type(4))) float v4f;
typedef __attribute__((ext_vector_type(8))) float v8f;

// ---------------------------------------------------------------------------
// Kernel 1: S = A + A^T   (512x512 f32, 1 MB -> stays resident in L2)
// ---------------------------------------------------------------------------
__global__ __launch_bounds__(256) void build_S_kernel(const float* __restrict__ A,
                                                      float* __restrict__ S) {
    int idx = blockIdx.x * 256 + threadIdx.x;   // 0 .. 512*512-1
    int i = idx >> 9;        // row
    int j = idx & 511;       // col
    S[idx] = A[i * DIM + j] + A[j * DIM + i];
}

// ---------------------------------------------------------------------------
// Kernel 2: out[:, :512] = p + q @ S   (f32 WMMA 16x16x4, S staged in LDS)
//           out[:, 512:] = q           (fused copy)
//
// Block tile: M=256 x N=64 (8 waves stacked in M, each wave M=32 x N=64).
// LDS holds the full-K S tile (512 x 64 f32 = 128 KB) with k-pairs
// interleaved so each B fragment is one ds_load_b64.
// Grid: (BATCH/256, DIM/64) = (256, 8).
// ---------------------------------------------------------------------------
__global__ __launch_bounds__(256) void fused_pq_gemm_kernel(const float* __restrict__ pq,
                                                            const float* __restrict__ S,
                                                            float* __restrict__ out) {
    // sh[t*64 + n] = ( S[2t][nbase+n], S[2t+1][nbase+n] )   t=0..255, n=0..63
    __shared__ v2f sh[256 * 64];   // 128 KB

    const int tid  = threadIdx.x;
    const int lane = tid & 31;
    const int wave = tid >> 5;
    const int half = lane >> 4;   // 0: lanes 0-15, 1: lanes 16-31
    const int l16  = lane & 15;

    const int mbase = blockIdx.x * 256 + wave * 32;   // batch-row base of this wave
    const int nbase = blockIdx.y * 64;                // output-col base of this block

    // ---- cooperative stage of S tile into LDS (K-pair interleaved)
    #pragma unroll 4
    for (int it = 0; it < 64; ++it) {
        int pid = it * 256 + tid;     // 0..16383
        int t   = pid >> 6;           // k-pair index 0..255
        int n   = pid & 63;           // column within tile
        v2f val;
        val.x = S[(size_t)(2 * t)     * DIM + nbase + n];
        val.y = S[(size_t)(2 * t + 1) * DIM + nbase + n];
        sh[pid] = val;
    }

    // ---- init accumulators with the p fragment (C/D layout: M = v + 8*half, N = l16)
    v8f acc[2][4];
    #pragma unroll
    for (int sub = 0; sub < 2; ++sub) {
        #pragma unroll
        for (int j = 0; j < 4; ++j) {
            #pragma unroll
            for (int v = 0; v < 8; ++v) {
                int row = mbase + 16 * sub + v + 8 * half;
                int col = nbase + 16 * j + l16;
                acc[sub][j][v] = pq[(size_t)row * 1024 + col];   // p part
            }
        }
    }

    __syncthreads();

    // q rows for the A fragments of the two M-subtiles (A layout: lane->M, half->K+2)
    const float* q0 = pq + (size_t)(mbase +      l16) * 1024 + 512;
    const float* q1 = pq + (size_t)(mbase + 16 + l16) * 1024 + 512;
    const v2f*   bp = sh + l16;    // B fragment base (lane->N)

    // ---- K loop: native WMMA K=4 per step, 128 steps
    #pragma unroll 4
    for (int k0 = 0; k0 < DIM; k0 += 4) {
        const int k    = k0 + 2 * half;          // this half-wave's K base
        const int trow = (k0 >> 1) + half;       // interleaved k-pair row in LDS

        // A fragments: contiguous K-pair per lane -> single b64 load each
        v2f a0 = *(const v2f*)(q0 + k);
        v2f a1 = *(const v2f*)(q1 + k);

        // B fragments: one ds_load_b64 per N-subtile
        v2f b[4];
        #pragma unroll
        for (int j = 0; j < 4; ++j)
            b[j] = bp[trow * 64 + 16 * j];

        #pragma unroll
        for (int j = 0; j < 4; ++j) {
            acc[0][j] = __builtin_amdgcn_wmma_f32_16x16x4_f32(
                false, a0, false, b[j], (short)0, acc[0][j], false, false);
            acc[1][j] = __builtin_amdgcn_wmma_f32_16x16x4_f32(
                false, a1, false, b[j], (short)0, acc[1][j], false, false);
        }
    }

    // ---- epilogue: store p + q@S into out[:, :512]
    #pragma unroll
    for (int sub = 0; sub < 2; ++sub) {
        #pragma unroll
        for (int j = 0; j < 4; ++j) {
            #pragma unroll
            for (int v = 0; v < 8; ++v) {
                int row = mbase + 16 * sub + v + 8 * half;
                int col = nbase + 16 * j + l16;
                out[(size_t)row * 1024 + col] = acc[sub][j][v];
            }
        }
    }

    // ---- fused copy: out[:, 512:] = q over this wave's 32x64 footprint
    #pragma unroll 4
    for (int i = 0; i < 16; ++i) {
        int row = mbase + 2 * i + half;
        size_t off = (size_t)row * 1024 + 512 + nbase + 4 * l16;
        *(v4f*)(out + off) = *(const v4f*)(pq + off);
    }
}

// ---------------------------------------------------------------------------
extern "C" void kernel_launch(void* const* d_in, const int* in_sizes, int n_in,
                              void* d_out, int out_size, void* d_ws, size_t ws_size,
                              hipStream_t stream) {
    const float* pq = (const float*)d_in[0];   // (BATCH, 1024) f32
    const float* A  = (const float*)d_in[1];   // (512, 512)   f32
    float* out = (float*)d_out;                // (BATCH, 1024) f32
    float* S   = (float*)d_ws;                 // 512*512*4 = 1 MB scratch

    build_S_kernel<<<(DIM * DIM) / 256, 256, 0, stream>>>(A, S);

    dim3 grid(BATCH / 256, DIM / 64);          // (256, 8)
    fused_pq_gemm_kernel<<<grid, 256, 0, stream>>>(pq, S, out);
}